// PartitionedNormalization_70480413328182
// MI455X (gfx1250) — compile-verified
//
#include <hip/hip_runtime.h>

// Partitioned BatchNorm for MI455X (gfx1250).
// Pipeline: zero-stats -> WMMA one-hot segment-sum stats -> finalize scale/bias -> streaming apply.
// Memory-bound: ~768 MiB min traffic -> ~33 us at 23.3 TB/s HBM.

#define NDOM 8
#define FEAT 256
#define EPS_BN 1e-3f

typedef float v2f __attribute__((ext_vector_type(2)));
typedef float v8f __attribute__((ext_vector_type(8)));

// Workspace layout (in floats):
#define WS_SUMS   0      // [8 x 256]
#define WS_SUMSQ  2048   // [8 x 256]
#define WS_COUNTS 4096   // [8]
#define WS_SCALE  4352   // [8 x 256]
#define WS_BIAS   6400   // [8 x 256]
#define WS_ZERO_N 4104   // sums + sumsq + counts

__device__ __forceinline__ v8f wmma_f32_16x16x4(v2f a, v2f b, v8f c) {
  // 8 args: (neg_a, A, neg_b, B, c_mod, C, reuse_a, reuse_b)
  return __builtin_amdgcn_wmma_f32_16x16x4_f32(false, a, false, b, (short)0, c,
                                               false, false);
}

__global__ void pn_zero(float* __restrict__ ws, int n) {
  int i = blockIdx.x * blockDim.x + threadIdx.x;
  if (i < n) ws[i] = 0.0f;
}

// Stats: per-wave 32-feature slice, 4 samples per step, one-hot WMMA segment-sum.
// D = A(16x4 one-hot of domain ids) x B(4x16 features) + C.
// C rows 0..7 (lanes 0..15) carry the 8 real domains; rows 8..15 stay zero.
__global__ void __launch_bounds__(256) pn_stats(const float* __restrict__ x,
                                                const int* __restrict__ dom,
                                                float* __restrict__ sums,
                                                float* __restrict__ sumsq,
                                                float* __restrict__ counts,
                                                int samples_per_block) {
  __shared__ float hist[NDOM];
  const int tid = threadIdx.x;
  if (tid < NDOM) hist[tid] = 0.0f;
  __syncthreads();

  const int wave = tid >> 5;   // 0..7 -> feature slice [wave*32, wave*32+32)
  const int lane = tid & 31;
  const int half = lane >> 4;  // 0: K=0,1  1: K=2,3
  const int l16  = lane & 15;
  const int f0   = wave * 32;

  v8f cs0 = {}; v8f cs1 = {};  // sum accumulators, 2 feature tiles
  v8f cq0 = {}; v8f cq1 = {};  // sum-of-squares accumulators

  const int s_begin = blockIdx.x * samples_per_block;
  const int s_end   = s_begin + samples_per_block;  // samples_per_block % 4 == 0

  for (int s = s_begin; s < s_end; s += 4) {
    const int rowA = s + half * 2;  // this lane supplies K = half*2 and half*2+1
    const int rowB = rowA + 1;
    const int d0 = dom[rowA];
    const int d1 = dom[rowB];

    const float* pA = x + (size_t)rowA * FEAT + f0 + l16;
    const float* pB = x + (size_t)rowB * FEAT + f0 + l16;

    v2f b0, b1;
    b0.x = pA[0];   b0.y = pB[0];    // tile 0: features f0 + l16
    b1.x = pA[16];  b1.y = pB[16];   // tile 1: features f0 + 16 + l16

    v2f a;                            // one-hot rows: M = l16 (both halves)
    a.x = (d0 == l16) ? 1.0f : 0.0f;
    a.y = (d1 == l16) ? 1.0f : 0.0f;

    v2f q0 = b0 * b0;
    v2f q1 = b1 * b1;

    cs0 = wmma_f32_16x16x4(a, b0, cs0);
    cq0 = wmma_f32_16x16x4(a, q0, cq0);
    cs1 = wmma_f32_16x16x4(a, b1, cs1);
    cq1 = wmma_f32_16x16x4(a, q1, cq1);

    // Count each sample exactly once: wave 0, lanes 0 and 16 cover s+0..s+3.
    if (wave == 0 && l16 == 0) {
      atomicAdd(&hist[d0], 1.0f);
      atomicAdd(&hist[d1], 1.0f);
    }
  }

  // Flush: C VGPR r, lanes 0-15 -> M=r (domains 0..7); lanes 16-31 hold M=r+8 == 0.
  if (half == 0) {
    const int f = f0 + l16;
#pragma unroll
    for (int r = 0; r < 8; ++r) {
      atomicAdd(&sums [r * FEAT + f     ], cs0[r]);
      atomicAdd(&sums [r * FEAT + f + 16], cs1[r]);
      atomicAdd(&sumsq[r * FEAT + f     ], cq0[r]);
      atomicAdd(&sumsq[r * FEAT + f + 16], cq1[r]);
    }
  }

  __syncthreads();
  if (tid < NDOM) atomicAdd(&counts[tid], hist[tid]);
}

// Fold stats + gammas/betas into per-(domain,feature) affine tables.
__global__ void pn_finalize(const float* __restrict__ sums,
                            const float* __restrict__ sumsq,
                            const float* __restrict__ counts,
                            const float* __restrict__ gg,
                            const float* __restrict__ gb,
                            const float* __restrict__ dg,
                            const float* __restrict__ db,
                            float* __restrict__ scale,
                            float* __restrict__ bias) {
  const int f = threadIdx.x;  // FEAT threads
  const float GG = gg[0], GB = gb[0];
#pragma unroll
  for (int d = 0; d < NDOM; ++d) {
    float c = counts[d];
    c = (c < 1.0f) ? 1.0f : c;
    const float inv_c = 1.0f / c;
    const float m   = sums[d * FEAT + f] * inv_c;
    float var       = sumsq[d * FEAT + f] * inv_c - m * m;
    var = (var < 0.0f) ? 0.0f : var;
    const float iv  = rsqrtf(var + EPS_BN);
    const float sc  = GG * dg[d] * iv;
    scale[d * FEAT + f] = sc;
    bias [d * FEAT + f] = GB + db[d] - m * sc;
  }
}

// Streaming normalize: b128 global loads/stores, scale/bias tables in LDS.
__global__ void __launch_bounds__(256) pn_apply(const float* __restrict__ x,
                                                const int* __restrict__ dom,
                                                const float* __restrict__ scale,
                                                const float* __restrict__ bias,
                                                float* __restrict__ out,
                                                int rows_per_block) {
  __shared__ float ls[NDOM * FEAT];
  __shared__ float lb[NDOM * FEAT];
  for (int i = threadIdx.x; i < NDOM * FEAT; i += 256) {
    ls[i] = scale[i];
    lb[i] = bias[i];
  }
  __syncthreads();

  const int tid  = threadIdx.x;
  const int c0   = (tid & 63) * 4;  // 64 groups of 4 features = 256
  const int rsub = tid >> 6;        // 4 rows in flight per step
  const int rbase = blockIdx.x * rows_per_block;
  const int rlim  = rbase + rows_per_block;

  for (int r = rbase + rsub; r < rlim; r += 4) {
    const int d = dom[r];
    const float4 xv = *(const float4*)(x + (size_t)r * FEAT + c0);
    const float4 sv = *(const float4*)(&ls[d * FEAT + c0]);  // ds_load_b128
    const float4 bv = *(const float4*)(&lb[d * FEAT + c0]);
    float4 o;
    o.x = fmaf(xv.x, sv.x, bv.x);
    o.y = fmaf(xv.y, sv.y, bv.y);
    o.z = fmaf(xv.z, sv.z, bv.z);
    o.w = fmaf(xv.w, sv.w, bv.w);
    *(float4*)(out + (size_t)r * FEAT + c0) = o;
  }
}

extern "C" void kernel_launch(void* const* d_in, const int* in_sizes, int n_in,
                              void* d_out, int out_size, void* d_ws, size_t ws_size,
                              hipStream_t stream) {
  (void)n_in; (void)out_size; (void)ws_size;

  const float* x  = (const float*)d_in[0];  // features [B, 256] f32
  const int*   dm = (const int*)d_in[1];    // domain_types [B] i32
  const float* gg = (const float*)d_in[2];  // global_gamma [1]
  const float* gb = (const float*)d_in[3];  // global_beta  [1]
  const float* dg = (const float*)d_in[4];  // domain_gammas [8]
  const float* db = (const float*)d_in[5];  // domain_betas  [8]
  float* out = (float*)d_out;
  float* ws  = (float*)d_ws;

  const int B = in_sizes[1];  // 262144; assumed multiple of 1024

  pn_zero<<<(WS_ZERO_N + 255) / 256, 256, 0, stream>>>(ws, WS_ZERO_N);

  const int spb = 256;            // samples per stats block (multiple of 4)
  pn_stats<<<B / spb, 256, 0, stream>>>(x, dm, ws + WS_SUMS, ws + WS_SUMSQ,
                                        ws + WS_COUNTS, spb);

  pn_finalize<<<1, FEAT, 0, stream>>>(ws + WS_SUMS, ws + WS_SUMSQ, ws + WS_COUNTS,
                                      gg, gb, dg, db, ws + WS_SCALE, ws + WS_BIAS);

  const int rpb = 128;            // rows per apply block (multiple of 4)
  pn_apply<<<B / rpb, 256, 0, stream>>>(x, dm, ws + WS_SCALE, ws + WS_BIAS, out, rpb);
}